// GatedDeltaMixerHeaded_7103875907805
// MI455X (gfx1250) — compile-verified
//
#include <hip/hip_runtime.h>
#include <hip/hip_bf16.h>

#define DIM 1024
#define HNUM 16
#define DH 64
#define KW 4
#define BATCH 8
#define SEQ 2048

typedef __attribute__((ext_vector_type(2))) float v2f;
typedef __attribute__((ext_vector_type(8))) float v8f;

#if defined(__has_builtin)
#if __has_builtin(__builtin_amdgcn_global_load_async_to_lds_b128) && \
    __has_builtin(__builtin_amdgcn_s_wait_asynccnt)
#define HAVE_ASYNC_LDS 1
#endif
#endif
#ifndef HAVE_ASYNC_LDS
#define HAVE_ASYNC_LDS 0
#endif

// Builtin wants: (int __vector(4) addrspace(1)*, ... lds ..., imm offset, imm cpol)
typedef int v4i_t __attribute__((vector_size(16)));
typedef __attribute__((address_space(1))) v4i_t* gptr_v4i;
typedef __attribute__((address_space(3))) v4i_t* lptr_v4i;

// 16-byte global -> LDS copy. Async (ASYNCcnt-tracked, no VGPR round trip) on
// gfx1250 toolchains that expose the builtin; plain staged copy otherwise.
__device__ __forceinline__ void copy16_g2l(const float* gsrc, float* ldst) {
#if HAVE_ASYNC_LDS
    __builtin_amdgcn_global_load_async_to_lds_b128((gptr_v4i)gsrc, (lptr_v4i)ldst,
                                                   0, 0);
#else
    *(float4*)ldst = *(const float4*)gsrc;
#endif
}

__device__ __forceinline__ void g2l_wait_all() {
#if HAVE_ASYNC_LDS
    __builtin_amdgcn_s_wait_asynccnt(0);
#endif
}

__device__ __forceinline__ v8f wmma_f32_16x16x4(v2f a, v2f b, v8f c) {
    // 8 args: (neg_a, A, neg_b, B, c_mod, C, reuse_a, reuse_b)
    return __builtin_amdgcn_wmma_f32_16x16x4_f32(false, a, false, b, (short)0, c,
                                                 false, false);
}

__device__ __forceinline__ float silu_f(float x) {
    return x / (1.0f + __expf(-x));
}

// ---------------------------------------------------------------------------
// GEMM: Y[m][n] = act( sum_k X[m][k] * W[n][k] + bias[n] )
// X: (M x K) row-major, W: (Nout x K) row-major (so B = W^T is K-inner).
// Block: 256 threads (8 waves), tiles 64x64 of Y. K staged in LDS chunks of 32
// with DOUBLE BUFFERING: chunk c+1 is copied (async-to-LDS) while the 16
// fp32 WMMAs of chunk c execute. Each wave: 16x16 in M, two 16-tiles in N.
// ---------------------------------------------------------------------------
__global__ __launch_bounds__(256)
void gemm_nt_wmma_f32(const float* __restrict__ X, const float* __restrict__ W,
                      const float* __restrict__ bias, float* __restrict__ Y,
                      int M, int Nn, int Kk, int act /*0:none 1:silu*/) {
    __shared__ float As[2][64 * 40];   // stride 40 floats: 16B-aligned rows
    __shared__ float Bs[2][64 * 40];

    const int tid  = threadIdx.x;
    const int wave = tid >> 5;
    const int lane = tid & 31;
    const int half = lane >> 4;        // 0: lanes 0-15, 1: lanes 16-31
    const int lm   = lane & 15;
    const int wm   = wave & 3;         // 4 waves along M
    const int wn   = wave >> 2;        // 2 waves along N

    const int m0 = blockIdx.x * 64;
    const int n0 = blockIdx.y * 64;

    // Per-thread tile-staging coordinates: 2 x 16B per thread per matrix.
    const int r0  = tid >> 3;               // 0..31
    const int c40 = (tid & 7) << 2;         // 0,4,...,28
    const int r1  = r0 + 32;                // 32..63

    v8f acc0 = {};
    v8f acc1 = {};

    auto load_tile = [&](int k0, int buf) {
        copy16_g2l(X + (size_t)(m0 + r0) * Kk + k0 + c40, &As[buf][r0 * 40 + c40]);
        copy16_g2l(X + (size_t)(m0 + r1) * Kk + k0 + c40, &As[buf][r1 * 40 + c40]);
        copy16_g2l(W + (size_t)(n0 + r0) * Kk + k0 + c40, &Bs[buf][r0 * 40 + c40]);
        copy16_g2l(W + (size_t)(n0 + r1) * Kk + k0 + c40, &Bs[buf][r1 * 40 + c40]);
    };

    load_tile(0, 0);
    g2l_wait_all();
    __syncthreads();

    int buf = 0;
    for (int k0 = 0; k0 < Kk; k0 += 32) {
        // Prefetch next chunk into the other buffer while we compute.
        if (k0 + 32 < Kk) load_tile(k0 + 32, buf ^ 1);

        const float* Ab = As[buf];
        const float* Bb = Bs[buf];
#pragma unroll
        for (int kk = 0; kk < 32; kk += 4) {
            // A frag: lane holds A[m][kk+2*half], A[m][kk+2*half+1]
            v2f a  = *(const v2f*)(Ab + (wm * 16 + lm) * 40 + kk + 2 * half);
            // B frag: lane holds B[kk+2*half][n], B[kk+2*half+1][n]; Bs is [n][k]
            v2f b0 = *(const v2f*)(Bb + (wn * 32 + lm) * 40 + kk + 2 * half);
            v2f b1 = *(const v2f*)(Bb + (wn * 32 + 16 + lm) * 40 + kk + 2 * half);
            acc0 = wmma_f32_16x16x4(a, b0, acc0);
            acc1 = wmma_f32_16x16x4(a, b1, acc1);
        }

        g2l_wait_all();     // next-chunk copies landed in LDS
        __syncthreads();    // all waves done reading current buffer
        buf ^= 1;
    }

    // Epilogue: C/D layout -> VGPR r holds row (r + 8*half), col = lane&15.
#pragma unroll
    for (int r = 0; r < 8; ++r) {
        int m  = m0 + wm * 16 + r + 8 * half;
        int n  = n0 + wn * 32 + lm;
        float y0 = acc0[r] + bias[n];
        float y1 = acc1[r] + bias[n + 16];
        if (act == 1) { y0 = silu_f(y0); y1 = silu_f(y1); }
        Y[(size_t)m * Nn + n]      = y0;
        Y[(size_t)m * Nn + n + 16] = y1;
    }
}

// ---------------------------------------------------------------------------
// Gate projections: a = sigmoid(x @ Wa^T + ba), b = sigmoid(x @ Wb^T + bb).
// One block per (b,t) row; 32 output dots (16 a + 16 b) split 8 ways over K.
// ---------------------------------------------------------------------------
__global__ __launch_bounds__(256)
void gates_kernel(const float* __restrict__ X, const float* __restrict__ Wa,
                  const float* __restrict__ ba, const float* __restrict__ Wb,
                  const float* __restrict__ bb, float* __restrict__ Aout,
                  float* __restrict__ Bout) {
    __shared__ float sred[256];
    const size_t row = blockIdx.x;
    const float* x = X + row * DIM;
    const int tid = threadIdx.x;
    const int hh  = tid & 31;      // 0..15 -> Wa, 16..31 -> Wb
    const int seg = tid >> 5;      // 8 K-segments of 128
    const float* w = (hh < HNUM) ? (Wa + (size_t)hh * DIM)
                                 : (Wb + (size_t)(hh - HNUM) * DIM);
    float s = 0.0f;
    const int kbeg = seg * 128;
#pragma unroll 4
    for (int k = kbeg; k < kbeg + 128; ++k) s += x[k] * w[k];
    sred[tid] = s;
    __syncthreads();
    if (tid < 32) {
        float tot = 0.0f;
#pragma unroll
        for (int j = 0; j < 8; ++j) tot += sred[tid + 32 * j];
        float bias = (hh < HNUM) ? ba[hh] : bb[hh - HNUM];
        float g = 1.0f / (1.0f + __expf(-(tot + bias)));
        if (hh < HNUM) Aout[row * HNUM + hh] = g;
        else           Bout[row * HNUM + (hh - HNUM)] = g;
    }
}

// ---------------------------------------------------------------------------
// Depthwise conv (K=4, pad=2, cross-correlation, outputs t in [0,N)) + SiLU.
// grid = (SEQ/256, DIM/256, B); thread owns one channel, slides over 256 t's.
// ---------------------------------------------------------------------------
__global__ __launch_bounds__(256)
void dwconv_silu(const float* __restrict__ U, const float* __restrict__ Wc,
                 float* __restrict__ Y) {
    const int c  = blockIdx.y * 256 + threadIdx.x;
    const int b  = blockIdx.z;
    const int t0 = blockIdx.x * 256;
    const float w0 = Wc[c * KW + 0], w1 = Wc[c * KW + 1];
    const float w2 = Wc[c * KW + 2], w3 = Wc[c * KW + 3];
    const float* Ub = U + (size_t)b * SEQ * DIM + c;
    float*       Yb = Y + (size_t)b * SEQ * DIM + c;
    auto ld = [&](int t) -> float {
        return (t >= 0 && t < SEQ) ? Ub[(size_t)t * DIM] : 0.0f;
    };
    float um2 = ld(t0 - 2), um1 = ld(t0 - 1), u0 = ld(t0);
    for (int t = t0; t < t0 + 256; ++t) {
        float up1 = ld(t + 1);
        float y = w0 * um2 + w1 * um1 + w2 * u0 + w3 * up1;
        Yb[(size_t)t * DIM] = silu_f(y);
        um2 = um1; um1 = u0; u0 = up1;
    }
}

// ---------------------------------------------------------------------------
// Row-wise l2norm over DIM, in place: x /= (||x|| + 1e-6). One block per row.
// ---------------------------------------------------------------------------
__global__ __launch_bounds__(256)
void l2norm_rows(float* __restrict__ X) {
    __shared__ float sred[256];
    float* p = X + (size_t)blockIdx.x * DIM;
    const int tid = threadIdx.x;
    float v[4];
    float ss = 0.0f;
#pragma unroll
    for (int i = 0; i < 4; ++i) {
        v[i] = p[tid + 256 * i];
        ss += v[i] * v[i];
    }
    sred[tid] = ss;
    __syncthreads();
    for (int s = 128; s > 0; s >>= 1) {
        if (tid < s) sred[tid] += sred[tid + s];
        __syncthreads();
    }
    const float scale = 1.0f / (sqrtf(sred[0]) + 1e-6f);
#pragma unroll
    for (int i = 0; i < 4; ++i) p[tid + 256 * i] = v[i] * scale;
}

// ---------------------------------------------------------------------------
// Gated delta-rule scan. One 64-thread block per (b,h). Thread e owns
// T[e][d] = S[d][e] (state transposed) so o[e] = sum_d q[d]*S[d][e] is a
// thread-local dot. Sk[d] = sum_e k[e]*S[d][e] needs the transposed matvec:
// stage k[e]*T[e][:] rows into padded LDS (stride 65 -> conflict-free column
// sums), thread d sums its column, broadcast Sk through LDS.
// ---------------------------------------------------------------------------
__global__ __launch_bounds__(64)
void delta_scan(const float* __restrict__ Q, const float* __restrict__ Kc,
                const float* __restrict__ V, const float* __restrict__ Ag,
                const float* __restrict__ Bg, float* __restrict__ O) {
    __shared__ float red[64 * 65];
    __shared__ float q_lds[64];
    __shared__ float v_lds[64];
    __shared__ float sk_lds[64];

    const int e = threadIdx.x;          // 0..63
    const int h = blockIdx.x;
    const int b = blockIdx.y;

    float T[DH];
#pragma unroll
    for (int d = 0; d < DH; ++d) T[d] = 0.0f;

    for (int t = 0; t < SEQ; ++t) {
        const size_t base  = ((size_t)b * SEQ + t) * DIM + h * DH;
        const size_t gbase = ((size_t)b * SEQ + t) * HNUM + h;
        const float ke = Kc[base + e];
        q_lds[e] = Q[base + e];
        v_lds[e] = V[base + e];
        const float av = Ag[gbase];
        const float bv = Bg[gbase];

#pragma unroll
        for (int d = 0; d < DH; ++d) red[e * 65 + d] = ke * T[d];
        __syncthreads();

        float sk = 0.0f;                // Sk[e] via column sum
#pragma unroll
        for (int d = 0; d < DH; ++d) sk += red[d * 65 + e];
        sk_lds[e] = sk;
        __syncthreads();

        const float ab = av * bv;
        float o = 0.0f;
#pragma unroll
        for (int d = 0; d < DH; ++d) {
            const float u = bv * v_lds[d] - ab * sk_lds[d];
            T[d] = av * T[d] + ke * u;  // S_new = a*S + (b*v - a*b*Sk) k^T
            o += q_lds[d] * T[d];       // o[e] = sum_d q[d] * S_new[d][e]
        }
        O[base + e] = o;
        __syncthreads();                // protect q_lds/v_lds/red for next t
    }
}

// ---------------------------------------------------------------------------
// Launch
// ---------------------------------------------------------------------------
extern "C" void kernel_launch(void* const* d_in, const int* in_sizes, int n_in,
                              void* d_out, int out_size, void* d_ws, size_t ws_size,
                              hipStream_t stream) {
    const float* x      = (const float*)d_in[0];
    const float* Wq     = (const float*)d_in[1];
    const float* bq     = (const float*)d_in[2];
    const float* Wk     = (const float*)d_in[3];
    const float* bk     = (const float*)d_in[4];
    const float* Wv     = (const float*)d_in[5];
    const float* bv     = (const float*)d_in[6];
    const float* Wa     = (const float*)d_in[7];
    const float* ba     = (const float*)d_in[8];
    const float* Wb     = (const float*)d_in[9];
    const float* bb     = (const float*)d_in[10];
    const float* conv_q = (const float*)d_in[11];
    const float* conv_k = (const float*)d_in[12];
    const float* conv_v = (const float*)d_in[13];
    const float* Wo     = (const float*)d_in[14];
    const float* bo     = (const float*)d_in[15];
    float* out = (float*)d_out;

    const size_t BIG = (size_t)BATCH * SEQ * DIM;     // 16.7M floats
    float* bufA = (float*)d_ws;                       // qpre -> kconv
    float* bufB = bufA + BIG;                         // kpre -> vconv
    float* bufC = bufB + BIG;                         // vpre -> O
    float* bufD = bufC + BIG;                         // qconv
    float* ag   = bufD + BIG;                         // (B*N*H)
    float* bg   = ag + (size_t)BATCH * SEQ * HNUM;

    const int M = BATCH * SEQ;                        // 16384 rows
    dim3 gemmGrid(M / 64, DIM / 64);                  // 256 x 16
    dim3 convGrid(SEQ / 256, DIM / 256, BATCH);       // 8 x 4 x 8

    // 1) q/k/v projections with fused SiLU (fp32 WMMA GEMMs, async-LDS pipe)
    gemm_nt_wmma_f32<<<gemmGrid, 256, 0, stream>>>(x, Wq, bq, bufA, M, DIM, DIM, 1);
    gemm_nt_wmma_f32<<<gemmGrid, 256, 0, stream>>>(x, Wk, bk, bufB, M, DIM, DIM, 1);
    gemm_nt_wmma_f32<<<gemmGrid, 256, 0, stream>>>(x, Wv, bv, bufC, M, DIM, DIM, 1);

    // 2) gate projections (sigmoid)
    gates_kernel<<<M, 256, 0, stream>>>(x, Wa, ba, Wb, bb, ag, bg);

    // 3) depthwise conv + SiLU, then l2norm for q/k (buffers recycled)
    dwconv_silu<<<convGrid, 256, 0, stream>>>(bufA, conv_q, bufD);  // q -> D
    l2norm_rows<<<M, 256, 0, stream>>>(bufD);
    dwconv_silu<<<convGrid, 256, 0, stream>>>(bufB, conv_k, bufA);  // k -> A
    l2norm_rows<<<M, 256, 0, stream>>>(bufA);
    dwconv_silu<<<convGrid, 256, 0, stream>>>(bufC, conv_v, bufB);  // v -> B

    // 4) gated delta-rule scan: O -> bufC
    delta_scan<<<dim3(HNUM, BATCH), 64, 0, stream>>>(bufD, bufA, bufB, ag, bg, bufC);

    // 5) output projection
    gemm_nt_wmma_f32<<<gemmGrid, 256, 0, stream>>>(bufC, Wo, bo, out, M, DIM, DIM, 0);
}